// MarginRanked_11802570130168
// MI455X (gfx1250) — compile-verified
//
#include <hip/hip_runtime.h>
#include <math.h>

typedef __attribute__((ext_vector_type(2))) float v2f;
typedef __attribute__((ext_vector_type(8))) float v8f;

namespace {
constexpr int kB   = 256;   // batch
constexpr int kC   = 1024;  // channels (K)
constexpr int kS2  = 225;   // 15*15 spatial (M and N)
constexpr int kNT  = 15;    // ceil(225/16) tiles
constexpr int kPad = 240;   // padded M extent for the LDS A panel
constexpr int kKC  = 16;    // K-chunk staged in LDS per pass (8 K-pairs)
}

// One block per (batch, pair). pair0: (cam[b], rad[b]) -> anchorsim + att
//                              pair1: (cam[cam_idx[b]], rad[b]) -> iimpsim
//                              pair2: (cam[b], rad[rad_idx[b]]) -> aimpsim
__global__ __launch_bounds__(256) void matchmax_kernel(
    const float* __restrict__ cam, const float* __restrict__ rad,
    const int* __restrict__ cam_idx, const int* __restrict__ rad_idx,
    float* __restrict__ sims,   // [3][256] in d_ws
    float* __restrict__ att)    // [256*225] (d_out + 1)
{
  const int b    = blockIdx.x;
  const int pair = blockIdx.y;
  int cb = b, rb = b;
  if (pair == 1) cb = cam_idx[b];
  if (pair == 2) rb = rad_idx[b];

  const float* __restrict__ Am = cam + (size_t)cb * kC * kS2;  // [K=1024, 225]
  const float* __restrict__ Bm = rad + (size_t)rb * kC * kS2;  // [K=1024, 225]

  // A panel, K-pair interleaved: Atile[((row>>1)*kPad + m)*2 + (row&1)].
  // A lane's WMMA fragment {A[krow][m], A[krow+1][m]} is one aligned b64.
  __shared__ float Atile[kKC * kPad];  // 15360 B
  __shared__ float colmax[kS2];
  __shared__ float red[256];

  const int tid  = threadIdx.x;
  const int wave = tid >> 5;   // 0..7
  const int lane = tid & 31;
  const int half = lane >> 4;  // 0: lanes 0-15, 1: lanes 16-31
  const int l16  = lane & 15;

  // Each wave owns N-tiles (2w, 2w+1); wave 7 duplicates tile 14 (uniform EXEC).
  const int nt0 = (2 * wave     < kNT) ? 2 * wave     : kNT - 1;
  const int nt1 = (2 * wave + 1 < kNT) ? 2 * wave + 1 : kNT - 1;
  const int ncol0 = nt0 * 16 + l16;
  const int ncol1 = nt1 * 16 + l16;
  const int nc0 = (ncol0 < kS2) ? ncol0 : kS2 - 1;  // clamp OOB column loads
  const int nc1 = (ncol1 < kS2) ? ncol1 : kS2 - 1;
  float cmax0 = -INFINITY, cmax1 = -INFINITY;

  // 15 M-tiles in 3 passes of 5; 5x2 accumulator block per wave (80 VGPRs).
  for (int g = 0; g < 3; ++g) {
    v8f acc[5][2];
    int mrow2[5];  // 2*clamped M row (dword index into interleaved panel)
#pragma unroll
    for (int t = 0; t < 5; ++t) {
      acc[t][0] = (v8f){0.f,0.f,0.f,0.f,0.f,0.f,0.f,0.f};
      acc[t][1] = (v8f){0.f,0.f,0.f,0.f,0.f,0.f,0.f,0.f};
      const int m = (g * 5 + t) * 16 + l16;
      mrow2[t] = ((m < kS2) ? m : kS2 - 1) * 2;       // clamp OOB row loads
    }

    for (int kc = 0; kc < kC; kc += kKC) {
      __syncthreads();
      // Cooperative fill: coalesced global reads, K-pair interleaved LDS store
      // (8-byte lane stride -> 32 distinct banks, conflict-free).
#pragma unroll
      for (int i = 0; i < (kKC * kPad) / 256; ++i) {  // 15 per thread
        const int idx  = tid + i * 256;
        const int row  = idx / kPad;
        const int col  = idx - row * kPad;
        const int colc = (col < kS2) ? col : kS2 - 1;
        const size_t goff = (size_t)(kc + row) * kS2 + colc;
        Atile[(((row >> 1) * kPad) + col) * 2 + (row & 1)] = Am[goff];
        if (kc + kKC < kC)  // speculative prefetch of next chunk
          __builtin_prefetch(&Am[goff + (size_t)kKC * kS2], 0, 1);
      }
      __syncthreads();

#pragma unroll
      for (int ks = 0; ks < kKC / 4; ++ks) {
        // f32 WMMA fragment layout: K = j + 2*half for VGPR j; M/N = lane&15.
        const int krow = kc + ks * 4 + (half << 1);
        const float* bp0 = Bm + (size_t)krow * kS2 + nc0;
        const float* bp1 = Bm + (size_t)krow * kS2 + nc1;
        v2f bf0; bf0.x = bp0[0]; bf0.y = bp0[kS2];
        v2f bf1; bf1.x = bp1[0]; bf1.y = bp1[kS2];
        const int pbase = (ks * 2 + half) * kPad * 2;  // K-pair row in panel
#pragma unroll
        for (int t = 0; t < 5; ++t) {
          const v2f af = *(const v2f*)&Atile[pbase + mrow2[t]];  // ds_load_b64
          acc[t][0] = __builtin_amdgcn_wmma_f32_16x16x4_f32(
              false, af, false, bf0, (short)0, acc[t][0], false, false);
          acc[t][1] = __builtin_amdgcn_wmma_f32_16x16x4_f32(
              false, af, false, bf1, (short)0, acc[t][1], false, false);
        }
      }
    }

    // Column max over valid rows: D VGPR r holds M = m0 + r + 8*half.
#pragma unroll
    for (int t = 0; t < 5; ++t) {
      const int m0 = (g * 5 + t) * 16;
#pragma unroll
      for (int r = 0; r < 8; ++r) {
        const int M = m0 + r + 8 * half;
        const float v0 = (M < kS2) ? acc[t][0][r] : -INFINITY;  // select only
        const float v1 = (M < kS2) ? acc[t][1][r] : -INFINITY;
        cmax0 = fmaxf(cmax0, v0);
        cmax1 = fmaxf(cmax1, v1);
      }
    }
  }

  // Merge the two half-wave M ranges (same column in lanes L and L+16).
  cmax0 = fmaxf(cmax0, __shfl_xor(cmax0, 16, 32));
  cmax1 = fmaxf(cmax1, __shfl_xor(cmax1, 16, 32));
  if (half == 0) {
    if (ncol0 < kS2) colmax[ncol0] = cmax0;
    if (ncol1 < kS2) colmax[ncol1] = cmax1;  // wave7 dup-writes identical value
  }
  __syncthreads();

  // Mean over the 225 columns; att = sigmoid(colmax) for the anchor pair.
  float v = (tid < kS2) ? colmax[tid] : 0.0f;
  if (pair == 0 && tid < kS2) {
    att[(size_t)b * kS2 + tid] = 1.0f / (1.0f + expf(-v));
  }
  red[tid] = v;
  __syncthreads();
  for (int s = 128; s > 0; s >>= 1) {
    if (tid < s) red[tid] += red[tid + s];
    __syncthreads();
  }
  if (tid == 0) sims[pair * kB + b] = red[0] / (float)kS2;
}

// Faithful replication of the stateful scan over b (sequential carry).
__global__ void loss_kernel(const float* __restrict__ sims, float* __restrict__ out)
{
  if (threadIdx.x == 0 && blockIdx.x == 0) {
    float prev = 0.0f, total = 0.0f;
    for (int i = 0; i < kB; ++i) {
      const float anchor = sims[0 * kB + i];
      const float iimp   = sims[1 * kB + i];
      const float aimp   = sims[2 * kB + i];
      const float a2i     = 1.0f + iimp - anchor;
      const float i2a_new = 1.0f + aimp - anchor;
      const float i2a = (a2i > 0.0f) ? i2a_new : prev;
      total += (a2i > 0.0f ? a2i : 0.0f) + (i2a > 0.0f ? i2a : 0.0f);
      prev = i2a;
    }
    out[0] = total / (float)kB;
  }
}

extern "C" void kernel_launch(void* const* d_in, const int* in_sizes, int n_in,
                              void* d_out, int out_size, void* d_ws, size_t ws_size,
                              hipStream_t stream) {
  const float* cam  = (const float*)d_in[0];
  const float* rad  = (const float*)d_in[1];
  const int*   cidx = (const int*)d_in[2];
  const int*   ridx = (const int*)d_in[3];
  float* out  = (float*)d_out;          // [0]=loss, [1..57600]=att
  float* sims = (float*)d_ws;           // 3*256 floats scratch

  dim3 grid(kB, 3), block(256);
  hipLaunchKernelGGL(matchmax_kernel, grid, block, 0, stream,
                     cam, rad, cidx, ridx, sims, out + 1);
  hipLaunchKernelGGL(loss_kernel, dim3(1), dim3(32), 0, stream, sims, out);
}